// Grouping_70781061038773
// MI455X (gfx1250) — compile-verified
//
#include <hip/hip_runtime.h>

typedef float v2f __attribute__((ext_vector_type(2)));
typedef float v8f __attribute__((ext_vector_type(8)));

// Problem shape (fixed by the reference).
constexpr int Bc = 16;
constexpr int Sc = 8192;
constexpr int Hc = 256;
constexpr int Gc = 1024;

constexpr int GT = 16;                      // groups per tile  (WMMA M)
constexpr int HT = 16;                      // h cols per wave  (WMMA N)
constexpr int WAVES = 4;                    // waves per block
constexpr int GTILES = Gc / GT;             // 64
constexpr int HBLOCKS = Hc / (HT * WAVES);  // 4

// ---------------------------------------------------------------------------
// Kernel 1: per-batch inclusive scan of group sizes -> starts[b, 0..G]
// starts[b, g] = sum(sizes[b, 0..g-1]);  starts[b, G] == S
// ---------------------------------------------------------------------------
__global__ __launch_bounds__(Gc) void grouping_scan_kernel(
    const int* __restrict__ sizes, int* __restrict__ starts) {
  __shared__ int sm[Gc];
  const int b = blockIdx.x;
  const int g = threadIdx.x;
  sm[g] = sizes[b * Gc + g];
  __syncthreads();
  for (int off = 1; off < Gc; off <<= 1) {
    int t = (g >= off) ? sm[g - off] : 0;
    __syncthreads();
    sm[g] += t;
    __syncthreads();
  }
  if (g == 0) starts[b * (Gc + 1)] = 0;
  starts[b * (Gc + 1) + g + 1] = sm[g];
}

// ---------------------------------------------------------------------------
// Kernel 2: segment-mean as one-hot matmul on the f32 WMMA pipe.
// One wave owns a 16-group x 16-column output tile and sweeps the contiguous
// S-range covering its 16 groups in K=8 steps (2x unrolled K=4 WMMAs):
//   D(16x16) += A(16x4 one-hot indicator) x B(4x16 feats slab)
// A-layout (ISA 7.12.2, 32-bit 16x4): lane m=lane&15 is row M=m;
//   lanes 0-15 hold K = {0,1} in {x,y}; lanes 16-31 hold K = {2,3}.
// C/D layout: VGPR i -> row M = i + 8*(lane>=16), col N = lane&15.
// All four B loads of a K=8 step are issued before the first WMMA so the
// second WMMA's operands stream in while the first executes.
// ---------------------------------------------------------------------------
__global__ __launch_bounds__(WAVES * 32) void grouping_mean_wmma_kernel(
    const float* __restrict__ feats, const int* __restrict__ sizes,
    const int* __restrict__ starts, float* __restrict__ out) {
  const int bid = blockIdx.x;
  const int hb = bid % HBLOCKS;
  const int gt = (bid / HBLOCKS) % GTILES;
  const int b = bid / (HBLOCKS * GTILES);

  const int lane = threadIdx.x & 31;
  const int wave = threadIdx.x >> 5;
  const int m = lane & 15;   // A row index, B/C column index
  const int hi = lane >> 4;  // half-wave select

  const int h0 = (hb * WAVES + wave) * HT;
  const int g0 = gt * GT;

  const int* __restrict__ st = starts + b * (Gc + 1);
  const int gs = st[g0 + m];      // start row of group (g0+m)
  const int ge = st[g0 + m + 1];  // end row (exclusive)
  const int sBeg = st[g0];        // first row covered by this group tile
  const int sEnd = st[g0 + GT];   // last row (exclusive)

  // Column base for this lane: feats[b, s, h0 + m] at stride Hc.
  const float* __restrict__ fcol = feats + ((size_t)b * Sc) * Hc + h0 + m;

  v8f acc = {};
  // Align the sweep to K=8; padding rows belong to neighboring group tiles,
  // so every indicator in this tile evaluates to 0 for them (self-masking).
  // sBeg&~7 >= 0 and (sEnd+7)&~7 <= S (S % 8 == 0): no OOB loads.
  const int sStop = (sEnd + 7) & ~7;
  for (int s = (sBeg & ~7); s < sStop; s += 8) {
    const int k0 = s + 2 * hi;  // this half-wave's K rows for step 0
    const int k1 = k0 + 4;      // ... and for step 1

    // Issue all four B loads first (software pipelining within the wave).
    v2f bv0, bv1;
    bv0.x = fcol[(size_t)k0 * Hc];
    bv0.y = fcol[(size_t)(k0 + 1) * Hc];
    bv1.x = fcol[(size_t)k1 * Hc];
    bv1.y = fcol[(size_t)(k1 + 1) * Hc];

    // Prefetch the K=8 step after next (global_prefetch_b8).
    __builtin_prefetch(fcol + (size_t)(k0 + 16) * Hc, 0, 1);

    v2f a0, a1;
    a0.x = (k0 >= gs && k0 < ge) ? 1.0f : 0.0f;
    a0.y = (k0 + 1 >= gs && k0 + 1 < ge) ? 1.0f : 0.0f;
    a1.x = (k1 >= gs && k1 < ge) ? 1.0f : 0.0f;
    a1.y = (k1 + 1 >= gs && k1 + 1 < ge) ? 1.0f : 0.0f;

    acc = __builtin_amdgcn_wmma_f32_16x16x4_f32(
        /*neg_a=*/false, a0, /*neg_b=*/false, bv0,
        /*c_mod=*/(short)0, acc, /*reuse_a=*/false, /*reuse_b=*/false);
    acc = __builtin_amdgcn_wmma_f32_16x16x4_f32(
        /*neg_a=*/false, a1, /*neg_b=*/false, bv1,
        /*c_mod=*/(short)0, acc, /*reuse_a=*/false, /*reuse_b=*/false);
  }

  // Scale rows by 1/size and store; each output element written exactly once.
  float* __restrict__ ocol = out + ((size_t)b * Gc + g0) * Hc + h0 + m;
#pragma unroll
  for (int i = 0; i < 8; ++i) {
    const int row = i + 8 * hi;
    const float denom = (float)sizes[b * Gc + g0 + row];  // sizes >= 1
    ocol[(size_t)row * Hc] = acc[i] / denom;
  }
}

// ---------------------------------------------------------------------------
extern "C" void kernel_launch(void* const* d_in, const int* in_sizes, int n_in,
                              void* d_out, int out_size, void* d_ws,
                              size_t ws_size, hipStream_t stream) {
  (void)in_sizes;
  (void)n_in;
  (void)out_size;
  (void)ws_size;

  const float* feats = (const float*)d_in[0];
  const int* sizes = (const int*)d_in[1];
  float* out = (float*)d_out;
  int* starts = (int*)d_ws;  // B * (G+1) ints = 65,600 bytes

  grouping_scan_kernel<<<Bc, Gc, 0, stream>>>(sizes, starts);

  dim3 grid(Bc * GTILES * HBLOCKS);
  dim3 block(WAVES * 32);
  grouping_mean_wmma_kernel<<<grid, block, 0, stream>>>(feats, sizes, starts,
                                                        out);
}